// DiffKSBase_62972810494173
// MI455X (gfx1250) — compile-verified
//
#include <hip/hip_runtime.h>
#include <stdint.h>

// DiffKS resonator scan for MI455X (gfx1250, wave32).
// B=16 batches (one workgroup each), N=16384 samples, K=108 lags, L=5.
// Min feedback delay z_center >= 37 > 32  =>  32-sample chunks are
// internally dependency-free: wave 0 computes a whole chunk per step.
// Waves 1..7 (plus wave 0's spare slots) precompute coefficients and the
// CDNA5 async-tensor path (global<->LDS async copies, ASYNCcnt) stages
// x in and drains y out, fully overlapped with the serial scan.

#define NB   16
#define NN   16384
#define SC   256          // super-chunk (samples per pipeline iteration)
#define NSC  (NN / SC)    // 64 iterations
#define CH   32           // chunk = wave32
#define RING 256          // y-history ring (max lookback 102+6 < 256)
#define RMASK (RING - 1)

// Low 32 bits of a generic pointer into LDS aperture == wave-relative LDS byte address
__device__ __forceinline__ unsigned lds_off(const void* p) {
  return (unsigned)(uintptr_t)p;
}

// GLOBAL_LOAD_ASYNC_TO_LDS_B32 (GVS mode): copies 4 bytes from global memory at saddr plus offset vaddr to the LDS address in vdst
__device__ __forceinline__ void async_ld_lds(unsigned ldsByte, unsigned voffBytes,
                                             const float* sbase) {
  asm volatile("global_load_async_to_lds_b32 %0, %1, %2"
               :: "v"(ldsByte), "v"(voffBytes), "s"(sbase) : "memory");
}
// GLOBAL_STORE_ASYNC_FROM_LDS_B32 (GVS mode): copies 4 bytes from the LDS address in vsrc to global memory at saddr plus offset vaddr
__device__ __forceinline__ void async_st_lds(unsigned voffBytes, unsigned ldsByte,
                                             float* dbase) {
  asm volatile("global_store_async_from_lds_b32 %0, %1, %2"
               :: "v"(voffBytes), "v"(ldsByte), "s"(dbase) : "memory");
}
__device__ __forceinline__ void wait_async0() {
  asm volatile("s_wait_asynccnt 0x0" ::: "memory");
}
__device__ __forceinline__ void wait_ds0() {
  asm volatile("s_wait_dscnt 0x0" ::: "memory");
}

// Per-sample coefficient math: taps, tuned period, Lagrange weights,
// convolved block coefficients, clamped integer delay base.
__device__ __forceinline__ void compute_coeffs(float f0v, float lb0, float lb1,
                                               float zHi, float* __restrict__ dst8) {
  float g  = 0.99f * lb0;
  float p  = lb1;
  float b0 = g * (1.0f - p);
  float a1 = g * p;
  float f0c = f0v - a1 / (b0 + a1 + 1e-7f);
  float zf  = floorf(f0c);
  float zc  = zf - 2.0f;                // z_center (float)
  float al  = f0c - zc;                 // alpha in [2,3)
  float u0 = al,        u1 = al - 1.0f, u2 = al - 2.0f;
  float u3 = al - 3.0f, u4 = al - 4.0f, u5 = al - 5.0f;
  // Lagrange weights, fixed denominators for L=5
  float w0 = (u1 * u2) * (u3 * u4) * u5 * (-1.0f / 120.0f);
  float w1 = (u0 * u2) * (u3 * u4) * u5 * ( 1.0f /  24.0f);
  float w2 = (u0 * u1) * (u3 * u4) * u5 * (-1.0f /  12.0f);
  float w3 = (u0 * u1) * (u2 * u4) * u5 * ( 1.0f /  12.0f);
  float w4 = (u0 * u1) * (u2 * u3) * u5 * (-1.0f /  24.0f);
  float w5 = (u0 * u1) * (u2 * u3) * u4 * ( 1.0f / 120.0f);
  // block = b0*[w,0] + a1*[0,w]  (stored positive; scan uses y = x + blk·hist)
  dst8[0] = b0 * w0;
  dst8[1] = fmaf(b0, w1, a1 * w0);
  dst8[2] = fmaf(b0, w2, a1 * w1);
  dst8[3] = fmaf(b0, w3, a1 * w2);
  dst8[4] = fmaf(b0, w4, a1 * w3);
  dst8[5] = fmaf(b0, w5, a1 * w4);
  dst8[6] = a1 * w5;
  // cols = zc-1+j clipped to [0,K-1] -> clamp base so cols stay in range
  dst8[7] = fminf(fmaxf(zc, 1.0f), zHi);
}

__global__ __launch_bounds__(SC) void diffks_scan_kernel(
    const float* __restrict__ f0, const float* __restrict__ x,
    const float* __restrict__ lb, const int* __restrict__ Kp,
    float* __restrict__ out) {
  __shared__ __align__(16) float yRing[RING];      // y history ring
  __shared__ __align__(16) float cBuf[2][SC * 8];  // double-buffered coeffs
  __shared__ __align__(16) float xBuf[2][SC];      // double-buffered x stage
  __shared__ __align__(16) float oBuf[2][SC];      // double-buffered out stage

  const int tid = threadIdx.x;
  const int b   = blockIdx.x;
  const long bbase = (long)b * NN;
  const float* xb  = x + bbase;
  float*       ob  = out + bbase;
  const float zHi  = (float)(*Kp) - 6.0f;          // K-6 = 102

  // ---- prologue: super-chunk 0 inputs (direct loads) ----
  {
    int s = tid;
    xBuf[0][tid] = xb[s];
    compute_coeffs(f0[bbase + s], lb[2 * (bbase + s)], lb[2 * (bbase + s) + 1],
                   zHi, &cBuf[0][tid * 8]);
  }
  __syncthreads();

  for (int t = 0; t < NSC; ++t) {
    const int nb = (t + 1) & 1;
    const int cb = t & 1;

    // All threads: prefetch next super-chunk (async x -> LDS, coeffs -> LDS)
    if (t + 1 < NSC) {
      int s = (t + 1) * SC + tid;
      async_ld_lds(lds_off(&xBuf[nb][tid]), (unsigned)(s * 4), xb);
      compute_coeffs(f0[bbase + s], lb[2 * (bbase + s)], lb[2 * (bbase + s) + 1],
                     zHi, &cBuf[nb][tid * 8]);
    }
    // All threads: drain previous super-chunk's outputs LDS -> HBM (async)
    if (t > 0) {
      int so = (t - 1) * SC + tid;
      async_st_lds((unsigned)(so * 4), lds_off(&oBuf[(t - 1) & 1][tid]), ob);
    }

    // Wave 0: serial scan of super-chunk t, 8 chunks of 32 samples
    if (tid < CH) {
      const float4* c4 = (const float4*)&cBuf[cb][0];
#pragma unroll
      for (int c = 0; c < SC / CH; ++c) {
        int ls = c * CH + tid;
        int n  = t * SC + ls;
        float4 k0 = c4[ls * 2 + 0];
        float4 k1 = c4[ls * 2 + 1];
        int zc = (int)k1.w;
        int i0 = n - zc;                 // nearest tap, >= 37 back
        float acc = xBuf[cb][ls];
        float yv;
        yv = (i0     >= 0) ? yRing[ i0      & RMASK] : 0.0f; acc = fmaf(k0.x, yv, acc);
        yv = (i0 - 1 >= 0) ? yRing[(i0 - 1) & RMASK] : 0.0f; acc = fmaf(k0.y, yv, acc);
        yv = (i0 - 2 >= 0) ? yRing[(i0 - 2) & RMASK] : 0.0f; acc = fmaf(k0.z, yv, acc);
        yv = (i0 - 3 >= 0) ? yRing[(i0 - 3) & RMASK] : 0.0f; acc = fmaf(k0.w, yv, acc);
        yv = (i0 - 4 >= 0) ? yRing[(i0 - 4) & RMASK] : 0.0f; acc = fmaf(k1.x, yv, acc);
        yv = (i0 - 5 >= 0) ? yRing[(i0 - 5) & RMASK] : 0.0f; acc = fmaf(k1.y, yv, acc);
        yv = (i0 - 6 >= 0) ? yRing[(i0 - 6) & RMASK] : 0.0f; acc = fmaf(k1.z, yv, acc);
        yRing[n & RMASK] = acc;
        oBuf[cb][ls]     = acc;
        wait_ds0();                      // order ring store vs next chunk's gathers
      }
    }

    wait_async0();                       // per-wave: async LDS writes landed
    __syncthreads();                     // publish cBuf/xBuf/yRing/oBuf
  }

  // ---- epilogue: drain last super-chunk ----
  {
    int so = (NSC - 1) * SC + tid;
    async_st_lds((unsigned)(so * 4), lds_off(&oBuf[(NSC - 1) & 1][tid]), ob);
    wait_async0();
  }
}

extern "C" void kernel_launch(void* const* d_in, const int* in_sizes, int n_in,
                              void* d_out, int out_size, void* d_ws, size_t ws_size,
                              hipStream_t stream) {
  (void)in_sizes; (void)n_in; (void)d_ws; (void)ws_size; (void)out_size;
  const float* f0 = (const float*)d_in[0];
  const float* x  = (const float*)d_in[1];
  const float* lb = (const float*)d_in[2];
  const int*   Kp = (const int*)d_in[3];
  float* out = (float*)d_out;
  hipLaunchKernelGGL(diffks_scan_kernel, dim3(NB), dim3(SC), 0, stream,
                     f0, x, lb, Kp, out);
}